// SPAVAE_3212635538240
// MI455X (gfx1250) — compile-verified
//
#include <hip/hip_runtime.h>
#include <hip/hip_bf16.h>

// ---------------- problem constants ----------------
#define BATCH     4096
#define INP       8000
#define H0DIM     128
#define H1DIM     64
#define LDIM      10
#define MI        400          // inducing points (20x20 grid)
#define JIT       1e-5f
#define RATIO     (40000.0f / 4096.0f)
#define LOG2PI    1.8378770664093453f

typedef float v2f __attribute__((ext_vector_type(2)));
typedef float v8f __attribute__((ext_vector_type(8)));

// ---------------- helpers ----------------
__device__ __forceinline__ float block_reduce(float v) {
  __shared__ float sm[256];
  __syncthreads();
  sm[threadIdx.x] = v;
  __syncthreads();
  for (int s = (int)blockDim.x >> 1; s > 0; s >>= 1) {
    if ((int)threadIdx.x < s) sm[threadIdx.x] += sm[threadIdx.x + s];
    __syncthreads();
  }
  return sm[0];
}

// ---------------- WMMA GEMM: C = alpha * A(MxK) @ B(KxN) + bias ----------------
// Register-blocked: each wave owns an (MB*16) x (NB*16) C macro-tile built from
// V_WMMA_F32_16X16X4_F32. A/B fragments are reused NB/MB times. Loop-carried
// pointers (A += 4, B += 4N) keep the inner loop free of 64-bit address muls.
// Fragment layout per ISA 7.12.2 (f32):
//   A 16x4: lanes 0-15 -> (M=lane, K=k0+{0,1}); lanes 16-31 -> (M=lane-16, K=k0+{2,3})
//   B 4x16: lanes 0-15 -> (K=k0+{0,1}, N=lane); lanes 16-31 -> (K=k0+{2,3}, N=lane-16)
//   C/D: VGPR r, lane<16 -> (M=r, N=lane); lane>=16 -> (M=r+8, N=lane-16)
template <int MB, int NB>
__global__ __launch_bounds__(256) void k_gemm_nn_t(const float* __restrict__ A,
    const float* __restrict__ Bm, const float* __restrict__ bias,
    float* __restrict__ C, int M, int N, int K, float alpha) {
  const int lane = threadIdx.x & 31;
  const int wave = threadIdx.x >> 5;
  const int ntn = N / (16 * NB);
  const int total = (M / (16 * MB)) * ntn;
  const int tile = blockIdx.x * 8 + wave;
  if (tile >= total) return;               // wave-uniform: EXEC stays all-1s for WMMA
  const int tm = (tile / ntn) * (16 * MB);
  const int tn = (tile % ntn) * (16 * NB);
  const int half = lane >> 4;              // 0 or 1
  const int kh = half << 1;                // 0 or 2
  const int mrow = lane & 15;

  const float* Ap[MB];
#pragma unroll
  for (int i = 0; i < MB; ++i) Ap[i] = A + (size_t)(tm + 16 * i + mrow) * K + kh;
  const float* Bp[NB];
#pragma unroll
  for (int j = 0; j < NB; ++j) Bp[j] = Bm + (size_t)kh * N + tn + 16 * j + mrow;
  const size_t bstep = (size_t)4 * N;

  const v8f vz = {0.f, 0.f, 0.f, 0.f, 0.f, 0.f, 0.f, 0.f};
  v8f acc[MB][NB];
#pragma unroll
  for (int i = 0; i < MB; ++i)
#pragma unroll
    for (int j = 0; j < NB; ++j) acc[i][j] = vz;

  for (int k0 = 0; k0 < K; k0 += 4) {
    v2f a[MB], b[NB];
#pragma unroll
    for (int i = 0; i < MB; ++i) { a[i].x = Ap[i][0]; a[i].y = Ap[i][1]; }
#pragma unroll
    for (int j = 0; j < NB; ++j) { b[j].x = Bp[j][0]; b[j].y = Bp[j][N]; }
#pragma unroll
    for (int i = 0; i < MB; ++i)
#pragma unroll
      for (int j = 0; j < NB; ++j)
        acc[i][j] = __builtin_amdgcn_wmma_f32_16x16x4_f32(false, a[i], false, b[j],
                                                          (short)0, acc[i][j],
                                                          false, false);
    // warm near caches for the streaming A operand; prefetch is speculative, so
    // no end-of-buffer guard is needed (stray prefetches are dropped).
#pragma unroll
    for (int i = 0; i < MB; ++i) __builtin_prefetch(Ap[i] + 64, 0, 3);
#pragma unroll
    for (int i = 0; i < MB; ++i) Ap[i] += 4;
#pragma unroll
    for (int j = 0; j < NB; ++j) Bp[j] += bstep;
  }

#pragma unroll
  for (int i = 0; i < MB; ++i) {
    const int rbase = tm + 16 * i + (half << 3);
#pragma unroll
    for (int j = 0; j < NB; ++j) {
      const int col = tn + 16 * j + mrow;
      const float bv = bias ? bias[col] : 0.f;
#pragma unroll
      for (int r = 0; r < 8; ++r)
        C[(size_t)(rbase + r) * N + col] = alpha * acc[i][j][r] + bv;
    }
  }
}

// ---------------- WMMA GEMM (TN): C = alpha * A^T diag(rs) B + D ----------------
// A is (K x M) row-major, B is (K x N) row-major.
// rs (length K) and D (M x N) are REQUIRED non-null: keeping them unconditional
// avoids per-iteration lane-masked guarded loads (saveexec/cndmask) in the hot loop.
__global__ __launch_bounds__(256) void k_gemm_tn(const float* __restrict__ A,
    const float* __restrict__ Bm, const float* __restrict__ D,
    const float* __restrict__ rs, float* __restrict__ C,
    int M, int N, int K, float alpha) {
  const int lane = threadIdx.x & 31;
  const int wave = threadIdx.x >> 5;
  const int ntn = N >> 4;
  const int total = (M >> 4) * ntn;
  const int tile = blockIdx.x * 8 + wave;
  if (tile >= total) return;
  const int tm = (tile / ntn) << 4;
  const int tn = (tile - (tile / ntn) * ntn) << 4;
  const int half = lane >> 4;
  const int kh = half << 1;
  const int am = tm + (lane & 15);
  const int bn = tn + (lane & 15);
  const float* pa = A + (size_t)kh * M + am;     // advance by 4*M per step
  const float* pb = Bm + (size_t)kh * N + bn;    // advance by 4*N per step
  const float* pr = rs + kh;                     // advance by 4 per step
  const size_t astep = (size_t)4 * M;
  const size_t bstep = (size_t)4 * N;
  v8f acc = {0.f, 0.f, 0.f, 0.f, 0.f, 0.f, 0.f, 0.f};
  for (int k0 = 0; k0 < K; k0 += 4) {
    v2f a, b;
    a.x = pa[0] * pr[0];
    a.y = pa[M] * pr[1];
    b.x = pb[0];
    b.y = pb[N];
    acc = __builtin_amdgcn_wmma_f32_16x16x4_f32(false, a, false, b, (short)0, acc,
                                                false, false);
    __builtin_prefetch(pa + 4 * astep, 0, 3);
    __builtin_prefetch(pb + 4 * bstep, 0, 3);
    pa += astep;
    pb += bstep;
    pr += 4;
  }
  const int rbase = tm + (half << 3);
  const int col = tn + (lane & 15);
#pragma unroll
  for (int r = 0; r < 8; ++r) {
    const size_t idx = (size_t)(rbase + r) * N + col;
    C[idx] = alpha * acc[r] + D[idx];
  }
}

// ---------------- BatchNorm stats over batch (per column) ----------------
__global__ __launch_bounds__(256) void k_bn_stats(const float* __restrict__ h, int C,
    float* __restrict__ mu, float* __restrict__ var) {
  const int col = blockIdx.x;
  float s = 0.f, s2 = 0.f;
  for (int r = threadIdx.x; r < BATCH; r += 256) {
    const float v = h[(size_t)r * C + col];
    s += v; s2 += v * v;
  }
  const float ts = block_reduce(s);
  const float ts2 = block_reduce(s2);
  if (threadIdx.x == 0) {
    const float m = ts * (1.0f / BATCH);
    mu[col] = m;
    var[col] = ts2 * (1.0f / BATCH) - m * m;
  }
}

__global__ void k_bn_apply(float* __restrict__ h, int C,
    const float* __restrict__ mu, const float* __restrict__ var,
    const float* __restrict__ g, const float* __restrict__ be) {
  const int gid = blockIdx.x * 256 + threadIdx.x;
  if (gid >= BATCH * C) return;
  const int c = gid % C;
  const float x = (h[gid] - mu[c]) * rsqrtf(var[c] + 1e-5f) * g[c] + be[c];
  h[gid] = x > 0.f ? x : expm1f(x);
}

// ---------------- q-net heads (K=64, N=10: too small/ragged for WMMA) ----------------
__global__ void k_qnet(const float* __restrict__ h1, const float* __restrict__ Wmu,
    const float* __restrict__ bmu, const float* __restrict__ Wvar,
    const float* __restrict__ bvar, float* __restrict__ qm, float* __restrict__ qv) {
  const int gid = blockIdx.x * 256 + threadIdx.x;
  if (gid >= BATCH * LDIM) return;
  const int i = gid / LDIM, l = gid % LDIM;
  const float* hr = h1 + (size_t)i * H1DIM;
  float sm = bmu[l], sv = bvar[l];
  for (int k = 0; k < H1DIM; ++k) {
    const float hv = hr[k];
    sm += hv * Wmu[k * LDIM + l];
    sv += hv * Wvar[k * LDIM + l];
  }
  qm[gid] = sm;
  sv = fminf(fmaxf(sv, -15.f), 15.f);
  qv[gid] = expf(sv);
}

// ---------------- Cauchy kernel matrix ----------------
__global__ void k_cauchy(const float* __restrict__ P, const float* __restrict__ Q,
    float* __restrict__ out, int rows, int cols) {
  const int gid = blockIdx.x * 256 + threadIdx.x;
  if (gid >= rows * cols) return;
  const int i = gid / cols, j = gid - (gid / cols) * cols;
  const float dx = P[i * 2] - Q[j * 2];
  const float dy = P[i * 2 + 1] - Q[j * 2 + 1];
  out[gid] = 1.0f / (1.0f + (dx * dx + dy * dy) * 10.0f);   // /SCALE(0.1)
}

// ---------------- Gauss-Jordan inverse (SPD, +JIT on diag), logdet from pivots ----------------
__global__ __launch_bounds__(1024) void k_gauss_jordan(const float* __restrict__ A,
    float* __restrict__ W, float* __restrict__ Ainv, float* __restrict__ ldslot) {
  const int tid = threadIdx.x;
  const int BD = 1024;
  __shared__ float s_piv;
  __shared__ float fac[MI];
  for (int idx = tid; idx < MI * 2 * MI; idx += BD) {
    const int r = idx / (2 * MI), c = idx - r * (2 * MI);
    float v;
    if (c < MI) { v = A[r * MI + c]; if (c == r) v += JIT; }
    else v = (c - MI == r) ? 1.f : 0.f;
    W[idx] = v;
  }
  float ld = 0.f;
  for (int k = 0; k < MI; ++k) {
    __syncthreads();
    if (tid == 0) { s_piv = W[k * (2 * MI) + k]; }
    __syncthreads();
    if (tid == 0) ld += logf(s_piv);
    const float ip = 1.0f / s_piv;
    for (int c = tid; c < 2 * MI; c += BD) W[k * (2 * MI) + c] *= ip;
    __syncthreads();
    for (int r = tid; r < MI; r += BD) fac[r] = W[r * (2 * MI) + k];
    __syncthreads();
    for (int r = 0; r < MI; ++r) {
      if (r == k) continue;
      const float f = fac[r];
      for (int c = tid; c < 2 * MI; c += BD)
        W[r * (2 * MI) + c] -= f * W[k * (2 * MI) + c];
    }
  }
  __syncthreads();
  for (int idx = tid; idx < MI * MI; idx += BD) {
    const int r = idx / MI, c = idx - r * MI;
    Ainv[idx] = W[r * (2 * MI) + MI + c];
  }
  if (tid == 0 && ldslot) *ldslot = ld;
}

// ---------------- LU logdet only (A + JIT*I); == 2*sum(log diag chol) for SPD ----------------
__global__ __launch_bounds__(256) void k_lu_logdet(const float* __restrict__ A,
    float* __restrict__ W, float* __restrict__ ldslot) {
  const int tid = threadIdx.x;
  const int BD = 256;
  __shared__ float s_piv;
  for (int idx = tid; idx < MI * MI; idx += BD) {
    const int r = idx / MI, c = idx - r * MI;
    W[idx] = A[idx] + ((r == c) ? JIT : 0.f);
  }
  float ld = 0.f;
  for (int k = 0; k < MI; ++k) {
    __syncthreads();
    if (tid == 0) s_piv = W[k * MI + k];
    __syncthreads();
    if (tid == 0) ld += logf(s_piv);
    const float ip = 1.0f / s_piv;
    for (int r = k + 1 + tid; r < MI; r += BD) {
      const float f = W[r * MI + k] * ip;
      for (int c = k; c < MI; ++c) W[r * MI + c] -= f * W[k * MI + c];
    }
  }
  if (tid == 0) *ldslot = ld;
}

// ---------------- row-wise dot: out[i] = sum_j T[i,j]*U[i,j] ----------------
__global__ __launch_bounds__(256) void k_rowdot(const float* __restrict__ T,
    const float* __restrict__ U, float* __restrict__ out, int cols) {
  const int row = blockIdx.x;
  const float* t = T + (size_t)row * cols;
  const float* u = U + (size_t)row * cols;
  float s = 0.f;
  for (int j = threadIdx.x; j < cols; j += 256) s += t[j] * u[j];
  const float r = block_reduce(s);
  if (threadIdx.x == 0) out[row] = r;
}

__global__ void k_pinv(const float* __restrict__ qv, int l, float* __restrict__ p) {
  const int gid = blockIdx.x * 256 + threadIdx.x;
  if (gid >= BATCH) return;
  p[gid] = 1.0f / qv[gid * LDIM + l];
}

// out[i*ostride] = alpha * sum_j A[i,j]*x[j]
__global__ void k_gemv(const float* __restrict__ A, const float* __restrict__ x,
    float* __restrict__ out, int M, int N, float alpha, int ostride) {
  const int gid = blockIdx.x * 256 + threadIdx.x;
  if (gid >= M) return;
  const float* a = A + (size_t)gid * N;
  float s = 0.f;
  for (int j = 0; j < N; ++j) s += a[j] * x[j];
  out[(size_t)gid * ostride] = alpha * s;
}

// out[j] = sum_i A[i,j] * yl[i*ystride] * p[i]
__global__ void k_gemv_tw(const float* __restrict__ A, const float* __restrict__ yl,
    int ystride, const float* __restrict__ p, float* __restrict__ out,
    int rows, int cols) {
  const int j = blockIdx.x * 256 + threadIdx.x;
  if (j >= cols) return;
  float s = 0.f;
  for (int i = 0; i < rows; ++i)
    s += A[(size_t)i * cols + j] * yl[(size_t)i * ystride] * p[i];
  out[j] = s;
}

__global__ void k_bv(const float* __restrict__ dsig, const float* __restrict__ dq,
    float* __restrict__ gpv_l) {
  const int gid = blockIdx.x * 256 + threadIdx.x;
  if (gid >= BATCH) return;
  gpv_l[(size_t)gid * 2] = 1.0f + dsig[gid] - dq[gid];
}

__global__ __launch_bounds__(256) void k_trace(const float* __restrict__ M1,
    float* __restrict__ slot) {
  float s = 0.f;
  for (int k = threadIdx.x; k < MI; k += 256) s += M1[k * MI + k];
  const float r = block_reduce(s);
  if (threadIdx.x == 0) *slot = r;
}

__global__ __launch_bounds__(256) void k_dot(const float* __restrict__ a,
    const float* __restrict__ b, float* __restrict__ slot) {
  float s = 0.f;
  for (int k = threadIdx.x; k < MI; k += 256) s += a[k] * b[k];
  const float r = block_reduce(s);
  if (threadIdx.x == 0) *slot = r;
}

// e_i = p(1-diagQ) + p*rd3 + log(nl) + p*(yl-mean2)^2 ; log(nl) = -log(p)
__global__ __launch_bounds__(256) void k_l3_partial(const float* __restrict__ p,
    const float* __restrict__ diagQ, const float* __restrict__ rd3,
    const float* __restrict__ yl, int ystride, const float* __restrict__ mean2,
    float* __restrict__ part) {
  const int gid = blockIdx.x * 256 + threadIdx.x;
  float e = 0.f;
  if (gid < BATCH) {
    const float pi = p[gid];
    const float d = yl[(size_t)gid * ystride] - mean2[gid];
    e = pi * (1.f - diagQ[gid]) + pi * rd3[gid] - logf(pi) + pi * d * d;
  }
  const float r = block_reduce(e);
  if (threadIdx.x == 0) part[blockIdx.x] = r;
}

__global__ __launch_bounds__(256) void k_gpce_partial(const float* __restrict__ gpm,
    const float* __restrict__ gpv, const float* __restrict__ qm,
    const float* __restrict__ qv, float* __restrict__ part) {
  const int gid = blockIdx.x * 256 + threadIdx.x;
  float e = 0.f;
  if (gid < BATCH * 2) {
    const int i = gid >> 1, l = gid & 1;
    const float pm = gpm[gid], pv = gpv[gid];
    const float mu = qm[i * LDIM + l], var = qv[i * LDIM + l];
    const float dm = mu - pm;
    e = -0.5f * (LOG2PI + logf(pv) + (var + dm * dm) / pv);
  }
  const float r = block_reduce(e);
  if (threadIdx.x == 0) part[blockIdx.x] = r;
}

__global__ __launch_bounds__(256) void k_gauss_partial(const float* __restrict__ qm,
    const float* __restrict__ qv, float* __restrict__ part) {
  const int gid = blockIdx.x * 256 + threadIdx.x;
  float e = 0.f;
  if (gid < BATCH * 8) {
    const int i = gid >> 3, l = (gid & 7) + 2;
    const float mu = qm[i * LDIM + l], var = qv[i * LDIM + l];
    e = 0.69314718055994531f + fabsf(mu) + 0.5f * var - 0.5f - 0.5f * logf(var);
  }
  const float r = block_reduce(e);
  if (threadIdx.x == 0) part[blockIdx.x] = r;
}

__global__ __launch_bounds__(256) void k_recon_partial(const float* __restrict__ ms,
    const float* __restrict__ y, float* __restrict__ part) {
  float s = 0.f;
  const size_t n = (size_t)BATCH * INP;
  for (size_t idx = (size_t)blockIdx.x * 256 + threadIdx.x; idx < n;
       idx += (size_t)1024 * 256) {
    const float d = ms[idx] - y[idx];
    s += d * d;
  }
  const float r = block_reduce(s);
  if (threadIdx.x == 0) part[blockIdx.x] = r;
}

__global__ __launch_bounds__(256) void k_reduce_final(const float* __restrict__ part,
    int n, float* __restrict__ dst) {
  float s = 0.f;
  for (int i = threadIdx.x; i < n; i += 256) s += part[i];
  const float r = block_reduce(s);
  if (threadIdx.x == 0) *dst = r;
}

__global__ void k_latent(const float* __restrict__ gpm, const float* __restrict__ gpv,
    const float* __restrict__ qm, const float* __restrict__ qv,
    const float* __restrict__ eps, float* __restrict__ lat) {
  const int gid = blockIdx.x * 256 + threadIdx.x;
  if (gid >= BATCH * LDIM) return;
  const int i = gid / LDIM, l = gid % LDIM;
  float pm, pv;
  if (l < 2) { pm = gpm[i * 2 + l]; pv = gpv[i * 2 + l]; }
  else       { pm = qm[gid];        pv = qv[gid]; }
  lat[gid] = pm + sqrtf(pv) * eps[gid];
}

__global__ void k_dec0(const float* __restrict__ lat, const float* __restrict__ dW0,
    const float* __restrict__ db0, float* __restrict__ hd) {
  const int gid = blockIdx.x * 256 + threadIdx.x;
  if (gid >= BATCH * H0DIM) return;
  const int i = gid / H0DIM, c = gid - (gid / H0DIM) * H0DIM;
  float s = db0[c];
  for (int k = 0; k < LDIM; ++k) s += lat[i * LDIM + k] * dW0[k * H0DIM + c];
  hd[gid] = s;
}

// S: 0=logdetK 1,2=logdetS 3,4=trace 5,6=quad 7,8=L3sum 9=gpce 10=gauss 11=recon 12=scratch
__global__ void k_combine(const float* __restrict__ S, float* __restrict__ out) {
  const float logdetK = S[0];
  const float KL0 = 0.5f * (logdetK - S[1] - (float)MI + S[3] + S[5]);
  const float KL1 = 0.5f * (logdetK - S[2] - (float)MI + S[4] + S[6]);
  const float L30 = -0.5f * (S[7] + (float)BATCH * LOG2PI);
  const float L31 = -0.5f * (S[8] + (float)BATCH * LOG2PI);
  const float inside = (L30 + L31) - ((float)BATCH / 40000.0f) * (KL0 + KL1);
  const float gp_KL = S[9] - inside;
  const float gauss_KL = S[10];
  const float recon = 100.0f * S[11];
  const float elbo = recon + 10.0f * (gp_KL + gauss_KL);
  out[0] = elbo; out[1] = recon; out[2] = gp_KL; out[3] = gauss_KL;
}

// ------------------------------- host driver -------------------------------
static inline int gblk(int M, int N, int MB, int NB) {
  return ((M / (16 * MB)) * (N / (16 * NB)) + 7) / 8;
}

extern "C" void kernel_launch(void* const* d_in, const int* in_sizes, int n_in,
                              void* d_out_v, int out_size, void* d_ws, size_t ws_size,
                              hipStream_t stream) {
  (void)in_sizes; (void)n_in; (void)out_size; (void)ws_size;
  const float* x    = (const float*)d_in[0];
  const float* y    = (const float*)d_in[1];
  const float* eps  = (const float*)d_in[2];
  const float* Z    = (const float*)d_in[3];
  const float* eW0  = (const float*)d_in[4];
  const float* eb0  = (const float*)d_in[5];
  const float* eg0  = (const float*)d_in[6];
  const float* ebe0 = (const float*)d_in[7];
  const float* eW1  = (const float*)d_in[8];
  const float* eb1  = (const float*)d_in[9];
  const float* eg1  = (const float*)d_in[10];
  const float* ebe1 = (const float*)d_in[11];
  const float* Wmu  = (const float*)d_in[12];
  const float* bmu  = (const float*)d_in[13];
  const float* Wvar = (const float*)d_in[14];
  const float* bvar = (const float*)d_in[15];
  const float* dW0  = (const float*)d_in[16];
  const float* db0  = (const float*)d_in[17];
  const float* dg0  = (const float*)d_in[18];
  const float* dbe0 = (const float*)d_in[19];
  const float* Wout = (const float*)d_in[20];
  const float* bout = (const float*)d_in[21];

  float* out = (float*)d_out_v;
  float* qm  = out + 4;
  float* qv  = out + 4 + (size_t)BATCH * LDIM;
  float* ms  = out + 4 + 2 * (size_t)BATCH * LDIM;
  float* lat = ms + (size_t)BATCH * INP;

  float* ws = (float*)d_ws;
  float* h0    = ws;                       // 4096*128 (reused as hd for decoder)
  float* h1    = h0 + 524288;              // 4096*64
  float* mu_   = h1 + 262144;              // 128
  float* var_  = mu_ + 128;                // 128
  float* Kmm   = var_ + 128;               // 400*400
  float* Kinv  = Kmm + 160000;             // 400*400
  float* GJ    = Kinv + 160000;            // 400*800 scratch (also LU scratch)
  float* Knm   = GJ + 320000;              // 4096*400
  float* Tb    = Knm + 1638400;            // 4096*400 (T1/T2/T3)
  float* diagQ = Tb + 1638400;             // 4096
  float* dsig  = diagQ + BATCH;            // 4096
  float* rd3   = dsig + BATCH;             // 4096
  float* pbuf  = rd3 + BATCH;              // 4096
  float* mean2 = pbuf + BATCH;             // 4096
  float* Sig   = mean2 + BATCH;            // 400*400
  float* Sinv  = Sig + 160000;             // 400*400
  float* KS    = Sinv + 160000;            // 400*400
  float* Ahat  = KS + 160000;              // 400*400
  float* M1    = Ahat + 160000;            // 400*400
  float* Qm    = M1 + 160000;              // 400*400
  float* vA    = Qm + 160000;              // 400 (Kmn_yp)
  float* vB    = vA + MI;                  // 400 (t)
  float* vC    = vB + MI;                  // 400 (mu_hat)
  float* vD    = vC + MI;                  // 400 (v)
  float* gpm   = vD + MI;                  // 4096*2
  float* gpv   = gpm + 2 * BATCH;          // 4096*2
  float* part  = gpv + 2 * BATCH;          // 2048 partials
  float* S     = part + 2048;              // 16 scalar slots

  // ---- encoder ----
  k_gemm_nn_t<2, 2><<<gblk(BATCH, H0DIM, 2, 2), 256, 0, stream>>>(y, eW0, eb0, h0,
      BATCH, H0DIM, INP, 1.f);
  k_bn_stats<<<H0DIM, 256, 0, stream>>>(h0, H0DIM, mu_, var_);
  k_bn_apply<<<(BATCH * H0DIM) / 256, 256, 0, stream>>>(h0, H0DIM, mu_, var_, eg0, ebe0);
  k_gemm_nn_t<2, 2><<<gblk(BATCH, H1DIM, 2, 2), 256, 0, stream>>>(h0, eW1, eb1, h1,
      BATCH, H1DIM, H0DIM, 1.f);
  k_bn_stats<<<H1DIM, 256, 0, stream>>>(h1, H1DIM, mu_, var_);
  k_bn_apply<<<(BATCH * H1DIM) / 256, 256, 0, stream>>>(h1, H1DIM, mu_, var_, eg1, ebe1);
  k_qnet<<<(BATCH * LDIM) / 256, 256, 0, stream>>>(h1, Wmu, bmu, Wvar, bvar, qm, qv);

  // ---- GP precomputation ----
  k_cauchy<<<(MI * MI + 255) / 256, 256, 0, stream>>>(Z, Z, Kmm, MI, MI);
  k_cauchy<<<(BATCH * MI + 255) / 256, 256, 0, stream>>>(x, Z, Knm, BATCH, MI);
  k_gauss_jordan<<<1, 1024, 0, stream>>>(Kmm, GJ, Kinv, S + 0);
  k_gemm_nn_t<2, 1><<<gblk(BATCH, MI, 2, 1), 256, 0, stream>>>(Knm, Kinv, nullptr, Tb,
      BATCH, MI, MI, 1.f);
  k_rowdot<<<BATCH, 256, 0, stream>>>(Tb, Knm, diagQ, MI);

  // ---- per GP latent dim ----
  for (int l = 0; l < 2; ++l) {
    k_pinv<<<BATCH / 256, 256, 0, stream>>>(qv, l, pbuf);
    k_gemm_tn<<<gblk(MI, MI, 1, 1), 256, 0, stream>>>(Knm, Knm, Kmm, pbuf, Sig,
        MI, MI, BATCH, RATIO);
    k_gauss_jordan<<<1, 1024, 0, stream>>>(Sig, GJ, Sinv, S + 12);
    k_gemv_tw<<<(MI + 255) / 256, 256, 0, stream>>>(Knm, qm + l, LDIM, pbuf, vA,
        BATCH, MI);
    k_gemv<<<(MI + 255) / 256, 256, 0, stream>>>(Sinv, vA, vB, MI, MI, 1.f, 1);
    k_gemv<<<BATCH / 256, 256, 0, stream>>>(Knm, vB, gpm + l, BATCH, MI, RATIO, 2);
    k_gemm_nn_t<2, 1><<<gblk(BATCH, MI, 2, 1), 256, 0, stream>>>(Knm, Sinv, nullptr,
        Tb, BATCH, MI, MI, 1.f);
    k_rowdot<<<BATCH, 256, 0, stream>>>(Tb, Knm, dsig, MI);
    k_bv<<<BATCH / 256, 256, 0, stream>>>(dsig, diagQ, gpv + l);
    k_gemv<<<(MI + 255) / 256, 256, 0, stream>>>(Kmm, vB, vC, MI, MI, RATIO, 1);
    k_gemm_nn_t<1, 1><<<gblk(MI, MI, 1, 1), 256, 0, stream>>>(Kmm, Sinv, nullptr, KS,
        MI, MI, MI, 1.f);
    k_gemm_nn_t<1, 1><<<gblk(MI, MI, 1, 1), 256, 0, stream>>>(KS, Kmm, nullptr, Ahat,
        MI, MI, MI, 1.f);
    k_lu_logdet<<<1, 256, 0, stream>>>(Ahat, GJ, S + 1 + l);
    k_gemm_nn_t<1, 1><<<gblk(MI, MI, 1, 1), 256, 0, stream>>>(Kinv, Ahat, nullptr, M1,
        MI, MI, MI, 1.f);
    k_trace<<<1, 256, 0, stream>>>(M1, S + 3 + l);
    k_gemv<<<(MI + 255) / 256, 256, 0, stream>>>(Kinv, vC, vD, MI, MI, 1.f, 1);
    k_dot<<<1, 256, 0, stream>>>(vC, vD, S + 5 + l);
    k_gemv<<<BATCH / 256, 256, 0, stream>>>(Knm, vD, mean2, BATCH, MI, 1.f, 1);
    k_gemm_nn_t<1, 1><<<gblk(MI, MI, 1, 1), 256, 0, stream>>>(M1, Kinv, nullptr, Qm,
        MI, MI, MI, 1.f);
    k_gemm_nn_t<2, 1><<<gblk(BATCH, MI, 2, 1), 256, 0, stream>>>(Knm, Qm, nullptr, Tb,
        BATCH, MI, MI, 1.f);
    k_rowdot<<<BATCH, 256, 0, stream>>>(Tb, Knm, rd3, MI);
    k_l3_partial<<<16, 256, 0, stream>>>(pbuf, diagQ, rd3, qm + l, LDIM, mean2, part);
    k_reduce_final<<<1, 256, 0, stream>>>(part, 16, S + 7 + l);
  }

  // ---- KL terms ----
  k_gpce_partial<<<32, 256, 0, stream>>>(gpm, gpv, qm, qv, part);
  k_reduce_final<<<1, 256, 0, stream>>>(part, 32, S + 9);
  k_gauss_partial<<<128, 256, 0, stream>>>(qm, qv, part);
  k_reduce_final<<<1, 256, 0, stream>>>(part, 128, S + 10);

  // ---- decoder ----
  k_latent<<<(BATCH * LDIM) / 256, 256, 0, stream>>>(gpm, gpv, qm, qv, eps, lat);
  k_dec0<<<(BATCH * H0DIM) / 256, 256, 0, stream>>>(lat, dW0, db0, h0);
  k_bn_stats<<<H0DIM, 256, 0, stream>>>(h0, H0DIM, mu_, var_);
  k_bn_apply<<<(BATCH * H0DIM) / 256, 256, 0, stream>>>(h0, H0DIM, mu_, var_, dg0, dbe0);
  k_gemm_nn_t<2, 2><<<gblk(BATCH, INP, 2, 2), 256, 0, stream>>>(h0, Wout, bout, ms,
      BATCH, INP, H0DIM, 1.f);
  k_recon_partial<<<1024, 256, 0, stream>>>(ms, y, part);
  k_reduce_final<<<1, 256, 0, stream>>>(part, 1024, S + 11);

  // ---- scalars ----
  k_combine<<<1, 1, 0, stream>>>(S, out);
}